// GAT_22436909154908
// MI455X (gfx1250) — compile-verified
//
#include <hip/hip_runtime.h>

typedef float v2f __attribute__((ext_vector_type(2)));
typedef float v8f __attribute__((ext_vector_type(8)));

#define NEG_SLOPE 0.2f

// ---------------------------------------------------------------------------
// WMMA GEMM: C[nrows, NC] = A[nrows, K] @ B[K, NC]   (all fp32, row-major)
// Block = 128 threads (4 waves). Each wave computes a 16-row x NCPAD-col tile
// with V_WMMA_F32_16X16X4_F32. B is staged in LDS (zero-padded to NCPAD cols).
// VGPR layouts per CDNA5 ISA:
//   A 16x4 f32 : lanes 0-15 hold K=k,k+1 ; lanes 16-31 hold K=k+2,k+3 (row = lane&15)
//   B 4x16 f32 : lanes 0-15 hold N=lane, rows k,k+1 ; lanes 16-31 N=lane-16, rows k+2,k+3
//   C 16x16 f32: vgpr v, lane L -> row = v + 8*(L>>4), col = L&15
// ---------------------------------------------------------------------------
template <int K, int NC, int NCPAD>
__launch_bounds__(128)
__global__ void gat_gemm_wmma(const float* __restrict__ A,
                              const float* __restrict__ B,
                              float* __restrict__ C, int nrows) {
  __shared__ float Bs[K * NCPAD];
  const int t = threadIdx.x;
  for (int idx = t; idx < K * NCPAD; idx += 128) {
    const int kk = idx / NCPAD, nn = idx % NCPAD;
    Bs[idx] = (nn < NC) ? B[kk * NC + nn] : 0.0f;
  }
  __syncthreads();

  const int wave = t >> 5;
  const int lane = t & 31;
  const int row0 = (blockIdx.x * 4 + wave) * 16;
  if (row0 >= nrows) return;  // wave-uniform: EXEC stays all-1s for live waves

  const int r = lane & 15;        // row-in-tile (A) / col-in-subtile (B,C)
  const int half = lane >> 4;     // 0 or 1
  const int koff = half * 2;      // which k-pair this lane holds

  constexpr int NT = NCPAD / 16;
  v8f acc[NT];
#pragma unroll
  for (int jt = 0; jt < NT; ++jt)
#pragma unroll
    for (int v = 0; v < 8; ++v) acc[jt][v] = 0.0f;

  const float* __restrict__ arow = A + (size_t)(row0 + r) * K;

  for (int k = 0; k < K; k += 4) {
    const int kk = k + koff;
    v2f a;
    a.x = arow[kk];
    a.y = arow[kk + 1];
#pragma unroll
    for (int jt = 0; jt < NT; ++jt) {
      const int col = jt * 16 + r;
      v2f b;
      b.x = Bs[kk * NCPAD + col];
      b.y = Bs[(kk + 1) * NCPAD + col];
      acc[jt] = __builtin_amdgcn_wmma_f32_16x16x4_f32(
          false, a, false, b, (short)0, acc[jt], false, false);
    }
  }

#pragma unroll
  for (int jt = 0; jt < NT; ++jt) {
    const int col = jt * 16 + r;
    if (col < NC) {
#pragma unroll
      for (int v = 0; v < 8; ++v) {
        const int rr = row0 + v + 8 * half;
        C[(size_t)rr * NC + col] = acc[jt][v];
      }
    }
  }
}

// ---------------------------------------------------------------------------
// Per-node attention logits: as[n] = h[n]·a_src ; ad[n] = h[n]·a_dst
// ---------------------------------------------------------------------------
template <int F>
__global__ void node_alpha(const float* __restrict__ h,
                           const float* __restrict__ a_s,
                           const float* __restrict__ a_d,
                           float* __restrict__ as, float* __restrict__ ad,
                           int N) {
  const int n = blockIdx.x * blockDim.x + threadIdx.x;
  if (n >= N) return;
  const float* __restrict__ hp = h + (size_t)n * F;
  float s = 0.f, d = 0.f;
#pragma unroll
  for (int f = 0; f < F; ++f) {
    const float hv = hp[f];
    s += hv * a_s[f];  // a_* are wave-uniform -> scalar loads
    d += hv * a_d[f];
  }
  as[n] = s;
  ad[n] = d;
}

__global__ void fill_f32(float* __restrict__ p, float v, long long n) {
  const long long i = blockIdx.x * (long long)blockDim.x + threadIdx.x;
  if (i < n) p[i] = v;
}

// int-punned float atomic max -> global_atomic_max_i32 / min_u32 (no CAS loop)
__device__ __forceinline__ void atomicMaxF32(float* addr, float val) {
  if (val >= 0.0f)
    atomicMax(reinterpret_cast<int*>(addr), __float_as_int(val));
  else
    atomicMin(reinterpret_cast<unsigned int*>(addr), __float_as_uint(val));
}

// Edge id >= E encodes the self-loop for node (id - E).
__device__ __forceinline__ void edge_endpoints(const long long* __restrict__ ei,
                                               int i, int E, int& s, int& d) {
  if (i < E) {
    s = (int)ei[i];
    d = (int)ei[(size_t)E + i];
  } else {
    s = d = i - E;
  }
}

// Pass 1: e = leaky_relu(as[src] + ad[dst]); segment max over dst
__global__ void edge_logits(const long long* __restrict__ ei,
                            const float* __restrict__ as,
                            const float* __restrict__ ad,
                            float* __restrict__ ebuf, float* __restrict__ m,
                            int E, int N) {
  const int i = blockIdx.x * blockDim.x + threadIdx.x;
  if (i >= E + N) return;
  int s, d;
  edge_endpoints(ei, i, E, s, d);
  float e = as[s] + ad[d];
  e = (e > 0.f) ? e : NEG_SLOPE * e;
  ebuf[i] = e;
  atomicMaxF32(&m[d], e);
}

// Pass 2: ex = exp(e - m[dst]); segment sum over dst (ebuf overwritten in place)
__global__ void edge_exp(const long long* __restrict__ ei,
                         const float* __restrict__ m, float* __restrict__ ebuf,
                         float* __restrict__ den, int E, int N) {
  const int i = blockIdx.x * blockDim.x + threadIdx.x;
  if (i >= E + N) return;
  int s, d;
  edge_endpoints(ei, i, E, s, d);
  const float ex = __expf(ebuf[i] - m[d]);
  ebuf[i] = ex;
  atomicAdd(&den[d], ex);
}

// Pass 3: out[dst] += (ex/den[dst]) * h[src]; F/4 threads per edge, float4 gathers
template <int F>
__global__ void edge_agg(const long long* __restrict__ ei,
                         const float* __restrict__ ebuf,
                         const float* __restrict__ den,
                         const float* __restrict__ h, float* __restrict__ out,
                         int E, int N) {
  constexpr int TPE = F / 4;
  const long long tid = blockIdx.x * (long long)blockDim.x + threadIdx.x;
  const long long tot = (long long)(E + N) * TPE;
  if (tid >= tot) return;
  const int i = (int)(tid / TPE);
  const int fo = (int)(tid % TPE) * 4;
  int s, d;
  edge_endpoints(ei, i, E, s, d);
  const float alpha = ebuf[i] / den[d];
  const float4 hv = *reinterpret_cast<const float4*>(h + (size_t)s * F + fo);
  float* op = out + (size_t)d * F + fo;
  atomicAdd(op + 0, alpha * hv.x);
  atomicAdd(op + 1, alpha * hv.y);
  atomicAdd(op + 2, alpha * hv.z);
  atomicAdd(op + 3, alpha * hv.w);
}

template <int F, bool RELU>
__global__ void bias_act(float* __restrict__ out, const float* __restrict__ b,
                         int N) {
  const long long i = blockIdx.x * (long long)blockDim.x + threadIdx.x;
  if (i >= (long long)N * F) return;
  float v = out[i] + b[(int)(i % F)];
  if (RELU) v = fmaxf(v, 0.f);
  out[i] = v;
}

// ---------------------------------------------------------------------------
extern "C" void kernel_launch(void* const* d_in, const int* in_sizes, int n_in,
                              void* d_out, int out_size, void* d_ws,
                              size_t ws_size, hipStream_t stream) {
  (void)n_in; (void)out_size; (void)ws_size;
  const float* x       = (const float*)d_in[0];
  const long long* ei  = (const long long*)d_in[1];  // int64 edge_index [2,E]
  const float* W1      = (const float*)d_in[2];
  const float* a1s     = (const float*)d_in[3];
  const float* a1d     = (const float*)d_in[4];
  const float* b1      = (const float*)d_in[5];
  const float* W2      = (const float*)d_in[6];
  const float* a2s     = (const float*)d_in[7];
  const float* a2d     = (const float*)d_in[8];
  const float* b2      = (const float*)d_in[9];
  float* out           = (float*)d_out;

  const int FIN = 256, H = 64, O = 20;
  const int N = in_sizes[0] / FIN;
  const int E = in_sizes[1] / 2;
  const long long Et = (long long)E + N;

  // workspace carve-up (all float, all 16B-aligned since N,E are mult. of 4)
  float* w    = (float*)d_ws;
  size_t o    = 0;
  float* h1   = w + o; o += (size_t)N * H;   // x @ W1
  float* agg1 = w + o; o += (size_t)N * H;   // layer-1 output (layer-2 input)
  float* h2   = w + o; o += (size_t)N * O;   // agg1 @ W2
  float* as   = w + o; o += N;               // reused across layers
  float* ad   = w + o; o += N;
  float* mbuf = w + o; o += N;
  float* den  = w + o; o += N;
  float* ebuf = w + o; o += (size_t)Et;

  const int T = 256;
  const int gN    = (N + T - 1) / T;
  const int gE    = (int)((Et + T - 1) / T);
  const int gemmB = (N / 16 + 3) / 4;  // 4 waves x 16 rows per block
  const int gNH   = (int)(((long long)N * H + T - 1) / T);
  const int gNO   = (int)(((long long)N * O + T - 1) / T);
  const int gAgg1 = (int)((Et * (H / 4) + T - 1) / T);
  const int gAgg2 = (int)((Et * (O / 4) + T - 1) / T);

  // ---------------- Layer 1 ----------------
  gat_gemm_wmma<256, 64, 64><<<gemmB, 128, 0, stream>>>(x, W1, h1, N);
  node_alpha<64><<<gN, T, 0, stream>>>(h1, a1s, a1d, as, ad, N);
  fill_f32<<<gN, T, 0, stream>>>(mbuf, -3.402823466e38f, N);
  fill_f32<<<gN, T, 0, stream>>>(den, 0.0f, N);
  fill_f32<<<gNH, T, 0, stream>>>(agg1, 0.0f, (long long)N * H);
  edge_logits<<<gE, T, 0, stream>>>(ei, as, ad, ebuf, mbuf, E, N);
  edge_exp<<<gE, T, 0, stream>>>(ei, mbuf, ebuf, den, E, N);
  edge_agg<64><<<gAgg1, T, 0, stream>>>(ei, ebuf, den, h1, agg1, E, N);
  bias_act<64, true><<<gNH, T, 0, stream>>>(agg1, b1, N);

  // ---------------- Layer 2 ----------------
  gat_gemm_wmma<64, 20, 32><<<gemmB, 128, 0, stream>>>(agg1, W2, h2, N);
  node_alpha<20><<<gN, T, 0, stream>>>(h2, a2s, a2d, as, ad, N);
  fill_f32<<<gN, T, 0, stream>>>(mbuf, -3.402823466e38f, N);
  fill_f32<<<gN, T, 0, stream>>>(den, 0.0f, N);
  fill_f32<<<gNO, T, 0, stream>>>(out, 0.0f, (long long)N * O);
  edge_logits<<<gE, T, 0, stream>>>(ei, as, ad, ebuf, mbuf, E, N);
  edge_exp<<<gE, T, 0, stream>>>(ei, mbuf, ebuf, den, E, N);
  edge_agg<20><<<gAgg2, T, 0, stream>>>(ei, ebuf, den, h2, out, E, N);
  bias_act<20, false><<<gNO, T, 0, stream>>>(out, b2, N);
}